// MambaMesh_17678085390974
// MI455X (gfx1250) — compile-verified
//
#include <hip/hip_runtime.h>

// ---------------------------------------------------------------------------
// MambaMesh random-walk + gather for MI455X (gfx1250).
//
// Shapes from reference setup: B=4, N=20000 faces, G=1024 walks/batch, L=63.
// Design (latency-bound on 64-step dependent walk chains; only 128 waves of
// walk work exist, so 1 wave32 per WGP is already max parallelism — optimize
// per-chain latency, everything into LDS):
//   * 1 block = 1 wave32 = 32 walks. grid = (G/32, B).
//   * TDM (tensor_load_to_lds) stages the batch neighbor table (240000 B)
//     into LDS once per block, overlapped with visited-bitset init and
//     global_prefetch of xyz. Walk steps chase pointers in LDS (~25 cy)
//     instead of L2 (~hundreds of cy).
//   * Per-lane visited bitset: ceil((N+1)/32)=626 dwords in LDS.
//   * Per-lane seq as u16 in LDS (no scratch anywhere); current face is
//     carried in a register so the hot chain never re-reads seq.
//     LDS total = 240000 + 32*626*4 + 32*64*2 = 324224 B <= 320 KB WGP LDS.
// ---------------------------------------------------------------------------

#define B_    4
#define N_    20000
#define G_    1024
#define LMAX  63
#define NBR_DW (N_ * 3)        // 60000 dwords = 240000 bytes
#define VIS_DW 626             // ceil((N_+1)/32)
#define TPB   32               // one wave32 per block

typedef unsigned int u32;
typedef unsigned long long u64;
typedef unsigned short u16;
typedef unsigned int u32x4 __attribute__((ext_vector_type(4)));
typedef unsigned int u32x8 __attribute__((ext_vector_type(8)));

// Deterministic counter-hash PRNG (PCG output fn). Seeded only from walk id,
// so every launch / graph replay produces identical output.
__device__ __forceinline__ u32 rng_next(u32& s) {
    s = s * 747796405u + 2891336453u;
    u32 w = ((s >> ((s >> 28u) + 4u)) ^ s) * 277803737u;
    return (w >> 22u) ^ w;
}

// uniform pick in [0, n) without integer divide: (r*n) >> 32
__device__ __forceinline__ u32 rng_range(u32& s, u32 n) {
    return (u32)(((u64)rng_next(s) * (u64)n) >> 32);
}

extern "C" __global__ __launch_bounds__(TPB)
void mambamesh_walk_gather(const float* __restrict__ xyz,
                           const int*   __restrict__ nbrs,
                           const int*   __restrict__ centers,
                           const int*   __restrict__ pN,
                           const int*   __restrict__ pL,
                           float*       __restrict__ out) {
    extern __shared__ u32 smem[];
    u32* s_nbr = smem;                               // [NBR_DW] TDM destination
    u32* s_vis = smem + NBR_DW;                      // [TPB][VIS_DW] visited bits
    u16* s_seq = (u16*)(smem + NBR_DW + TPB * VIS_DW); // [TPB][LMAX+1] walk seq

    const int tid = threadIdx.x;
    const int b   = blockIdx.y;
    const int g   = blockIdx.x * TPB + tid;

    int N = pN[0]; if (N > N_) N = N_;               // bitset sized for N_ faces
    int L = pL[0]; if (L > LMAX) L = LMAX;

    // ---- TDM: async DMA this batch's neighbor table (240 KB) into LDS ----
    {
        u64 gaddr = (u64)(const void*)(nbrs + (size_t)b * NBR_DW);
        u32 lds_off = __builtin_amdgcn_groupstaticsize(); // dynamic LDS base

        u32x4 g0;
        g0[0] = 1u;                                   // count=1, user mode, no gather
        g0[1] = lds_off;                              // lds_addr (bytes)
        g0[2] = (u32)gaddr;                           // global_addr[31:0]
        g0[3] = (u32)((gaddr >> 32) & 0x01FFFFFFu)    // global_addr[56:32]
              | (2u << 30);                           // type=2 ("image")

        u32x8 g1;
        g1[0] = 2u << 16;                             // wg_mask=0, data_size=4B
        g1[1] = ((u32)NBR_DW & 0xFFFFu) << 16;        // tensor_dim0[15:0]  @63:48
        g1[2] = ((u32)NBR_DW >> 16)                   // tensor_dim0[31:16] @79:64
              | (1u << 16);                           // tensor_dim1=1      @95:80
        g1[3] = ((u32)NBR_DW & 0xFFFFu) << 16;        // tile_dim0=60000    @127:112
        g1[4] = 0u;                                   // tile_dim1=0, tile_dim2=0 (1-D)
        g1[5] = (u32)NBR_DW;                          // tensor_dim0_stride[31:0]
        g1[6] = 0u;
        g1[7] = 0u;
        asm volatile("tensor_load_to_lds %0, %1" :: "s"(g0), "s"(g1) : "memory");
    }

    // ---- overlap with TDM: clear this lane's visited bitset ----
    u32* vis = s_vis + tid * VIS_DW;
    #pragma unroll 4
    for (int w = 0; w < VIS_DW; ++w) vis[w] = 0u;

    // ---- overlap with TDM: warm L2/WGP$ with this batch's xyz for gather ----
    {
        const char* base = (const char*)(xyz + (size_t)b * N_ * 3);
        for (int off = tid * 256; off < N_ * 3 * 4; off += TPB * 256)
            __builtin_prefetch(base + off, 0, 1);     // global_prefetch_b8
    }

    const int f0 = centers[(size_t)b * G_ + g];
    vis[(u32)N  >> 5] |= 1u << ((u32)N  & 31u);       // sentinel face n_faces
    vis[(u32)f0 >> 5] |= 1u << ((u32)f0 & 31u);

    u16* sq = s_seq + tid * (LMAX + 1);
    sq[0] = (u16)f0;

    u32 st = ((u32)(b * G_ + g) + 42u) * 0x9E3779B9u + 0x85EBCA6Bu;
    (void)rng_next(st);

    __builtin_amdgcn_s_wait_tensorcnt(0);             // neighbor table resident

    auto is_vis = [&](int f) -> bool {
        return (vis[(u32)f >> 5] >> ((u32)f & 31u)) & 1u;
    };

    // ---- the walk (mirrors reference while-loop + backtracking) ----
    int i = 1, bs = 1;
    int cur = f0;                                     // == seq[i-1], in-register
    while (i <= L) {
        const int a3 = cur * 3;
        const int n0 = (int)s_nbr[a3], n1 = (int)s_nbr[a3 + 1], n2 = (int)s_nbr[a3 + 2];

        int cand[3]; int cnt = 0;
        if (!is_vis(n0)) cand[cnt++] = n0;
        if (!is_vis(n1)) cand[cnt++] = n1;
        if (!is_vis(n2)) cand[cnt++] = n2;

        int to_add; bool hit = false; int bs_f = bs;
        if (cnt > 0) {
            to_add = cand[rng_range(st, (u32)cnt)];
        } else {
            // stuck: backtrack every other step until an unvisited branch
            bool found = false; int ta = 0; int bsc = bs;
            while (!found && i > bsc) {
                const int back = (int)sq[i - bsc - 1];
                const int bb   = back * 3;
                const int m0 = (int)s_nbr[bb], m1 = (int)s_nbr[bb + 1], m2 = (int)s_nbr[bb + 2];
                int bcand[3]; int bc = 0;
                if (!is_vis(m0)) bcand[bc++] = m0;
                if (!is_vis(m1)) bcand[bc++] = m1;
                if (!is_vis(m2)) bcand[bc++] = m2;
                const u32 pick = rng_range(st, (u32)(bc > 0 ? bc : 1));
                if (bc > 0) { found = true; ta = bcand[pick]; }
                else        { bsc += 2; }
            }
            if (found) { to_add = ta; hit = true; bs_f = bsc; }
            else       { to_add = (int)rng_range(st, (u32)N); }
        }

        int i_new;
        if (hit) {                        // rewind: overwrite seq[i-bs_f .. i)
            i_new = i - bs_f;
            for (int j = i_new; j < i; ++j) sq[j] = (u16)to_add;
            bs = bs_f;
        } else {
            i_new = i;
            bs = 1;
        }
        vis[(u32)to_add >> 5] |= 1u << ((u32)to_add & 31u);
        sq[i_new] = (u16)to_add;
        cur = to_add;                     // seq[i_new] == next iteration's seq[i-1]
        i = i_new + 1;
    }

    // ---- gather + recenter: out[b][g][k][:] = xyz[seq[k]] - xyz[f0] ----
    const float* px = xyz + (size_t)b * N_ * 3;
    const float cx = px[(size_t)f0 * 3 + 0];
    const float cy = px[(size_t)f0 * 3 + 1];
    const float cz = px[(size_t)f0 * 3 + 2];
    float* o = out + ((size_t)b * G_ + (size_t)g) * (size_t)(L + 1) * 3;
    for (int k = 0; k <= L; ++k) {
        const int f = (int)sq[k];
        o[(size_t)k * 3 + 0] = px[(size_t)f * 3 + 0] - cx;
        o[(size_t)k * 3 + 1] = px[(size_t)f * 3 + 1] - cy;
        o[(size_t)k * 3 + 2] = px[(size_t)f * 3 + 2] - cz;
    }
}

extern "C" void kernel_launch(void* const* d_in, const int* in_sizes, int n_in,
                              void* d_out, int out_size, void* d_ws, size_t ws_size,
                              hipStream_t stream) {
    (void)in_sizes; (void)n_in; (void)out_size; (void)d_ws; (void)ws_size;
    const float* xyz     = (const float*)d_in[0];  // [B,N,3] f32
    const int*   nbrs    = (const int*)  d_in[1];  // [B,N,3] i32
    const int*   centers = (const int*)  d_in[2];  // [B,G]   i32
    const int*   pN      = (const int*)  d_in[3];  // scalar n_faces
    const int*   pL      = (const int*)  d_in[4];  // scalar seq_len
    float*       out     = (float*)d_out;          // [B,G,L+1,3] f32

    dim3 grid(G_ / TPB, B_, 1);
    dim3 block(TPB, 1, 1);
    const size_t smem = (size_t)(NBR_DW + TPB * VIS_DW) * 4   // nbrs + visited
                      + (size_t)TPB * (LMAX + 1) * 2;          // seq (u16)
    hipLaunchKernelGGL(mambamesh_walk_gather, grid, block, smem, stream,
                       xyz, nbrs, centers, pN, pL, out);
}